// SparKEncoder_83236466196612
// MI455X (gfx1250) — compile-verified
//
#include <hip/hip_runtime.h>
#include <hip/hip_bf16.h>
#include <stdint.h>

// ---------------------------------------------------------------------------
// CDNA5 / gfx1250 sparse-conv encoder
//   stage0: conv3^3 s1  1->64   (VALU, bandwidth bound)
//   stage1: conv3^3 s2  64->128 (WMMA f32 16x16x4 implicit GEMM)
//   stage2: conv3^3 s2 128->256 (WMMA)
//   stage3: conv3^3 s2 256->768 (WMMA)
// each followed by masked-BN (deterministic 2-pass reduction) + ReLU + mask
// ---------------------------------------------------------------------------

typedef __attribute__((ext_vector_type(2))) float v2f;
typedef __attribute__((ext_vector_type(8))) float v8f;

static constexpr int NBLK_RED = 256;          // partial-reduction blocks
static constexpr int PSTRIDE  = 2 * 768 + 8;  // floats per partial record

// ----------------------------- stage 0 -------------------------------------
// block = 256 threads = 4 voxels x 64 channels
__global__ __launch_bounds__(256) void conv0_kernel(
    const float* __restrict__ x, const unsigned char* __restrict__ mask,
    const float* __restrict__ w, float* __restrict__ y, int B, int Dd)
{
  __shared__ float taps[4][27];
  const int tid = threadIdx.x;
  const int vs  = tid >> 6;        // voxel within block
  const int co  = tid & 63;        // output channel
  const int DHW = Dd * Dd * Dd;
  const int vi  = blockIdx.x * 4 + vs;

  const int b   = vi / DHW;
  int rem       = vi - b * DHW;
  const int od  = rem / (Dd * Dd); rem -= od * Dd * Dd;
  const int oh  = rem / Dd;
  const int ow  = rem - oh * Dd;

  if (co < 27) {
    const int kd = co / 9, kh = (co / 3) % 3, kw = co % 3;
    const int id = od - 1 + kd, ih = oh - 1 + kh, iw = ow - 1 + kw;
    float v = 0.f;
    if ((unsigned)id < (unsigned)Dd && (unsigned)ih < (unsigned)Dd &&
        (unsigned)iw < (unsigned)Dd) {
      const int idx = ((b * Dd + id) * Dd + ih) * Dd + iw;  // Cin == 1
      v = mask[idx] ? x[idx] : 0.f;
    }
    taps[vs][co] = v;
  }
  __syncthreads();

  float acc = 0.f;
#pragma unroll
  for (int k = 0; k < 27; ++k) acc = fmaf(taps[vs][k], w[k * 64 + co], acc);
  y[(size_t)vi * 64 + co] = acc;
}

// ------------------------ stages 1..3: WMMA conv ---------------------------
// block = 128 threads (4 x wave32); tile = 32 voxels (M) x 128 channels (N)
// wave (m,n) = (w>>1, w&1); each wave: 16(M) x 64(N) via four 16x16 accums,
// so each A fragment feeds 4 WMMAs (minimizes LDS reads per matrix op).
template <int CIN, int COUT>
__global__ __launch_bounds__(128) void conv_s2_wmma_kernel(
    const float* __restrict__ x, const float* __restrict__ w,
    float* __restrict__ y, int Di, int Do)
{
  constexpr int CK = 32;                 // Cin chunk
  __shared__ float As[32][CK + 4];       // [voxel-row][ci]   (stride 36: 16B)
  __shared__ float Bs[128][CK + 2];      // [n][k] transposed (stride 34: 8B)
  __shared__ int   rowBase[32];

  const int tid    = threadIdx.x;
  const int lane   = tid & 31;
  const int wv     = tid >> 5;
  const int wave_m = wv >> 1;            // 0..1 -> voxel rows 16*wave_m
  const int wave_n = wv & 1;             // 0..1 -> channels 64*wave_n
  const int vtile  = blockIdx.x * 32;
  const int ncol0  = blockIdx.y * 128;
  const int DoHW   = Do * Do;

  v8f acc[4];
#pragma unroll
  for (int j = 0; j < 4; ++j) acc[j] = (v8f){};

  // per-row output coords (threads 0..31 own one voxel row each)
  int ob = 0, od = 0, oh = 0, ow = 0;
  if (tid < 32) {
    const int vi = vtile + tid;
    ob = vi / (Do * DoHW);
    int rem = vi - ob * Do * DoHW;
    od = rem / DoHW; rem -= od * DoHW;
    oh = rem / Do;
    ow = rem - oh * Do;
  }

  for (int off = 0; off < 27; ++off) {
    const int kd = off / 9, kh = (off / 3) % 3, kw = off % 3;
    __syncthreads();
    if (tid < 32) {
      const int id = od * 2 - 1 + kd;
      const int ih = oh * 2 - 1 + kh;
      const int iw = ow * 2 - 1 + kw;
      const bool ok = (unsigned)id < (unsigned)Di &&
                      (unsigned)ih < (unsigned)Di &&
                      (unsigned)iw < (unsigned)Di;
      rowBase[tid] = ok ? (((ob * Di + id) * Di + ih) * Di + iw) * CIN : -1;
    }
    if (off + 1 < 27) {  // pull next tap's weights toward L2 (global_prefetch)
      __builtin_prefetch((const void*)(w + (size_t)((off + 1) * CIN) * COUT + ncol0), 0, 1);
    }
    __syncthreads();

    for (int cb = 0; cb < CIN; cb += CK) {
      { // A tile: 32 rows x 32 ci; each thread loads 8 contiguous channels
        const int row = tid & 31;
        const int c0  = (tid >> 5) * 8;
        const int base = rowBase[row];
        float4 v0 = make_float4(0.f, 0.f, 0.f, 0.f), v1 = v0;
        if (base >= 0) {
          const float4* p = (const float4*)(x + (size_t)base + cb + c0);
          v0 = p[0];
          v1 = p[1];
        }
        *(float4*)&As[row][c0]     = v0;
        *(float4*)&As[row][c0 + 4] = v1;
      }
      { // B tile: 32 k x 128 n, transposed into Bs[n][k]
        const int k  = tid & 31;
        const int n0 = (tid >> 5) * 32;
        const float* wp =
            w + (size_t)(off * CIN + cb + k) * COUT + ncol0 + n0;
#pragma unroll
        for (int j = 0; j < 32; ++j) Bs[n0 + j][k] = wp[j];
      }
      __syncthreads();

#pragma unroll
      for (int k4 = 0; k4 < CK; k4 += 4) {
        const int kb = k4 + ((lane >> 4) << 1);          // K pair per half-wave
        const int r  = lane & 15;
        const int nb = wave_n * 64 + r;
        v2f a = *(const v2f*)&As[wave_m * 16 + r][kb];
#pragma unroll
        for (int j = 0; j < 4; ++j) {
          v2f bf = *(const v2f*)&Bs[nb + j * 16][kb];
          acc[j] = __builtin_amdgcn_wmma_f32_16x16x4_f32(
              false, a, false, bf, (short)0, acc[j], false, false);
        }
      }
      __syncthreads();
    }
  }

  // writeback: C/D layout — lane<16: M=e, lane>=16: M=e+8; N = lane&15
  const int mrow = wave_m * 16 + (lane >> 4) * 8;
  const int nc0  = ncol0 + wave_n * 64 + (lane & 15);
  float* yp = y + (size_t)(vtile + mrow) * COUT + nc0;
#pragma unroll
  for (int j = 0; j < 4; ++j) {
#pragma unroll
    for (int e = 0; e < 8; ++e) {
      yp[(size_t)e * COUT + j * 16] = acc[j][e];
    }
  }
}

// -------------------------- mask any-pool 2^3 ------------------------------
__global__ void maskpool_kernel(const unsigned char* __restrict__ mi,
                                unsigned char* __restrict__ mo, int B, int Do)
{
  const int i = blockIdx.x * blockDim.x + threadIdx.x;
  const int n = B * Do * Do * Do;
  if (i >= n) return;
  const int Di = Do * 2;
  const int b  = i / (Do * Do * Do);
  int rem      = i - b * Do * Do * Do;
  const int od = rem / (Do * Do); rem -= od * Do * Do;
  const int oh = rem / Do;
  const int ow = rem - oh * Do;
  unsigned r = 0;
#pragma unroll
  for (int dz = 0; dz < 2; ++dz)
#pragma unroll
    for (int dy = 0; dy < 2; ++dy)
#pragma unroll
      for (int dx = 0; dx < 2; ++dx)
        r |= mi[((b * Di + 2 * od + dz) * Di + 2 * oh + dy) * Di + 2 * ow + dx];
  mo[i] = r ? 1 : 0;
}

// --------------------- masked BN: deterministic 2-pass ---------------------
template <int COUT>
__global__ __launch_bounds__(256) void bn_reduce_kernel(
    const float* __restrict__ y, const unsigned char* __restrict__ m,
    float* __restrict__ partials, int nvox)
{
  constexpr int SLOTS = (COUT + 255) / 256;
  float s[SLOTS], ss[SLOTS];
#pragma unroll
  for (int i = 0; i < SLOTS; ++i) { s[i] = 0.f; ss[i] = 0.f; }
  float cnt = 0.f;
  const int c = threadIdx.x;

  for (int v = blockIdx.x; v < nvox; v += gridDim.x) {
    const float mf = m[v] ? 1.f : 0.f;
    if (c == 0) cnt += mf;
#pragma unroll
    for (int si = 0; si < SLOTS; ++si) {
      const int cc = c + si * 256;
      if (cc < COUT) {
        const float val = y[(size_t)v * COUT + cc] * mf;
        s[si]  += val;
        ss[si] += val * val;
      }
    }
  }
  float* pb = partials + (size_t)blockIdx.x * PSTRIDE;
#pragma unroll
  for (int si = 0; si < SLOTS; ++si) {
    const int cc = c + si * 256;
    if (cc < COUT) { pb[cc] = s[si]; pb[COUT + cc] = ss[si]; }
  }
  if (c == 0) pb[2 * COUT] = cnt;
}

template <int COUT>
__global__ __launch_bounds__(256) void bn_finalize_kernel(
    const float* __restrict__ partials, const float* __restrict__ g,
    const float* __restrict__ b, float* __restrict__ scaleshift, int nblk)
{
  constexpr int SLOTS = (COUT + 255) / 256;
  __shared__ float s_cnt;
  if (threadIdx.x == 0) {
    float c = 0.f;
    for (int i = 0; i < nblk; ++i) c += partials[(size_t)i * PSTRIDE + 2 * COUT];
    s_cnt = fmaxf(c, 1.f);
  }
  __syncthreads();
  const float cnt = s_cnt;

#pragma unroll
  for (int si = 0; si < SLOTS; ++si) {
    const int cc = threadIdx.x + si * 256;
    if (cc < COUT) {
      float sum = 0.f, sq = 0.f;
      for (int i = 0; i < nblk; ++i) {
        sum += partials[(size_t)i * PSTRIDE + cc];
        sq  += partials[(size_t)i * PSTRIDE + COUT + cc];
      }
      const float mean = sum / cnt;
      const float var  = sq / cnt - mean * mean;
      const float sc   = g[cc] * rsqrtf(var + 1e-5f);
      scaleshift[cc]        = sc;
      scaleshift[COUT + cc] = b[cc] - mean * sc;
    }
  }
}

template <int COUT>
__global__ void bn_norm_kernel(float* __restrict__ y,
                               const unsigned char* __restrict__ m,
                               const float* __restrict__ ss, long long n)
{
  const long long i = (long long)blockIdx.x * blockDim.x + threadIdx.x;
  if (i >= n) return;
  const int c        = (int)(i % COUT);
  const long long v  = i / COUT;
  const float mf     = m[v] ? 1.f : 0.f;
  const float val    = fmaf(y[i], ss[c], ss[COUT + c]);
  y[i] = fmaxf(val, 0.f) * mf;
}

// ------------------------------- launcher ----------------------------------
extern "C" void kernel_launch(void* const* d_in, const int* in_sizes, int n_in,
                              void* d_out, int out_size, void* d_ws,
                              size_t ws_size, hipStream_t stream)
{
  (void)in_sizes; (void)n_in; (void)out_size; (void)ws_size;

  const float* x  = (const float*)d_in[0];
  const unsigned char* m0 = (const unsigned char*)d_in[1];
  const float* w0 = (const float*)d_in[2];
  const float* g0 = (const float*)d_in[3];
  const float* b0 = (const float*)d_in[4];
  const float* w1 = (const float*)d_in[5];
  const float* g1 = (const float*)d_in[6];
  const float* b1 = (const float*)d_in[7];
  const float* w2 = (const float*)d_in[8];
  const float* g2 = (const float*)d_in[9];
  const float* b2 = (const float*)d_in[10];
  const float* w3 = (const float*)d_in[11];
  const float* g3 = (const float*)d_in[12];
  const float* b3 = (const float*)d_in[13];

  // voxel counts per stage
  const int nv0 = 2 * 64 * 64 * 64;  // 524288
  const int nv1 = 2 * 32 * 32 * 32;  // 65536
  const int nv2 = 2 * 16 * 16 * 16;  // 8192
  const int nv3 = 2 * 8 * 8 * 8;     // 1024

  // d_out layout: final[2,8,8,8,768] | x0[.,64] | x1[.,128] | x2[.,256]
  float* out   = (float*)d_out;
  float* y3    = out;
  float* y0    = out + (size_t)nv3 * 768;                 // +786432
  float* y1    = y0 + (size_t)nv0 * 64;                   // +33554432
  float* y2    = y1 + (size_t)nv1 * 128;                  // +8388608

  // workspace layout
  char* wsb = (char*)d_ws;
  unsigned char* m1 = (unsigned char*)wsb;                //  65536 B
  unsigned char* m2 = m1 + 65536;                         //   8192 B
  unsigned char* m3 = m2 + 8192;                          //   1024 B
  float* partials   = (float*)(wsb + 131072);             // NBLK*PSTRIDE f32
  float* scsh       = (float*)(wsb + 131072 +
                               (size_t)NBLK_RED * PSTRIDE * sizeof(float));

  // ---- stage 0: conv 1->64 (s1) + BN(mask m0) ----
  conv0_kernel<<<nv0 / 4, 256, 0, stream>>>(x, m0, w0, y0, 2, 64);
  bn_reduce_kernel<64><<<NBLK_RED, 256, 0, stream>>>(y0, m0, partials, nv0);
  bn_finalize_kernel<64><<<1, 256, 0, stream>>>(partials, g0, b0, scsh, NBLK_RED);
  {
    const long long n = (long long)nv0 * 64;
    bn_norm_kernel<64><<<(unsigned)((n + 255) / 256), 256, 0, stream>>>(y0, m0, scsh, n);
  }

  // ---- stage 1: conv 64->128 (s2) + BN(mask m1) ----
  maskpool_kernel<<<(nv1 + 255) / 256, 256, 0, stream>>>(m0, m1, 2, 32);
  conv_s2_wmma_kernel<64, 128>
      <<<dim3(nv1 / 32, 128 / 128), 128, 0, stream>>>(y0, w1, y1, 64, 32);
  bn_reduce_kernel<128><<<NBLK_RED, 256, 0, stream>>>(y1, m1, partials, nv1);
  bn_finalize_kernel<128><<<1, 256, 0, stream>>>(partials, g1, b1, scsh, NBLK_RED);
  {
    const long long n = (long long)nv1 * 128;
    bn_norm_kernel<128><<<(unsigned)((n + 255) / 256), 256, 0, stream>>>(y1, m1, scsh, n);
  }

  // ---- stage 2: conv 128->256 (s2) + BN(mask m2) ----
  maskpool_kernel<<<(nv2 + 255) / 256, 256, 0, stream>>>(m1, m2, 2, 16);
  conv_s2_wmma_kernel<128, 256>
      <<<dim3(nv2 / 32, 256 / 128), 128, 0, stream>>>(y1, w2, y2, 32, 16);
  bn_reduce_kernel<256><<<NBLK_RED, 256, 0, stream>>>(y2, m2, partials, nv2);
  bn_finalize_kernel<256><<<1, 256, 0, stream>>>(partials, g2, b2, scsh, NBLK_RED);
  {
    const long long n = (long long)nv2 * 256;
    bn_norm_kernel<256><<<(unsigned)((n + 255) / 256), 256, 0, stream>>>(y2, m2, scsh, n);
  }

  // ---- stage 3: conv 256->768 (s2) + BN(mask m3) ----
  maskpool_kernel<<<(nv3 + 255) / 256, 256, 0, stream>>>(m2, m3, 2, 8);
  conv_s2_wmma_kernel<256, 768>
      <<<dim3(nv3 / 32, 768 / 128), 128, 0, stream>>>(y2, w3, y3, 16, 8);
  bn_reduce_kernel<768><<<NBLK_RED, 256, 0, stream>>>(y3, m3, partials, nv3);
  bn_finalize_kernel<768><<<1, 256, 0, stream>>>(partials, g3, b3, scsh, NBLK_RED);
  {
    const long long n = (long long)nv3 * 768;
    bn_norm_kernel<768><<<(unsigned)((n + 255) / 256), 256, 0, stream>>>(y3, m3, scsh, n);
  }
}